// ARIMA_74663711473750
// MI455X (gfx1250) — compile-verified
//
#include <hip/hip_runtime.h>

// ---------------- problem constants ----------------
#define S0_LEN   1048577            // raw series length
#define S_LEN    1048576            // differenced length (D=1)
#define PRE      17                 // t = 0..16: err_t = y_t (pred gated by t > m=16)
#define NP       1048576            // padded recurrence steps (t = 17 .. 17+NP-1)
#define NP_REAL  (S_LEN - PRE)      // 1048559 real steps in the recurrence
#define NCHUNK   4096               // parallel chunks
#define CSTEPS   256                // timesteps per chunk (= 16 blocks of 16)

// ---------------- workspace float offsets ----------------
#define ACC_OFF  0                  // 1   : sum of squared errors
#define S0_OFF   16                 // 16  : initial state (err_1..err_16)
#define GB_OFF   64                 // 256 : A^16, row-major
#define GC_OFF   320                // 256 : A^256
#define U_OFF    576                // 256 : [A^15 e | ... | A^0 e]
#define M_OFF    832                // 16*256 : M_j = GB^{15-j} U
#define V_OFF    8192               // NCHUNK*16 : per-chunk offset vectors
#define SIN_OFF  81920              // NCHUNK*16 : per-chunk incoming states
#define B_OFF    163840             // NP : b_t stream (t = p+17)
// total ~ 4.85 MB of f32 workspace

typedef __attribute__((ext_vector_type(2))) float v2f;
typedef __attribute__((ext_vector_type(8))) float v8f;

__device__ __forceinline__ v8f wmma4(v2f a, v2f b, v8f c) {
  // D(16x16) = A(16x4) x B(4x16) + C   -- chain 4 of these for 16x16x16
  return __builtin_amdgcn_wmma_f32_16x16x4_f32(
      /*neg_a=*/false, a, /*neg_b=*/false, b,
      /*c_mod=*/(short)0, c, /*reuse_a=*/false, /*reuse_b=*/false);
}

// ================= phase 1: scalar setup (1 block x 256) =================
__global__ void arima_setup(const float* __restrict__ series,
                            const float* __restrict__ w_ma,
                            float* __restrict__ ws) {
  __shared__ float A[16][16], T1[16][16], T2[16][16], GBsh[16][16], Ush[16][16];
  __shared__ float u[16], un[16], pre[32];
  const int tid = threadIdx.x;
  const int i = tid >> 4, k = tid & 15;

  // prelude: err_t = y_t for t = 0..16; state s0 = (err_1..err_16)
  if (tid < 32) pre[tid] = 0.f;
  if (tid < PRE) {
    float yt = series[tid + 1] - series[tid];
    pre[tid] = yt * yt;
    if (tid >= 1) ws[S0_OFF + tid - 1] = yt;
  }
  // companion matrix: rows 0..14 shift up, row 15 = -w_ma
  A[i][k] = (i < 15) ? ((k == i + 1) ? 1.f : 0.f) : -w_ma[k];
  __syncthreads();
  if (tid == 0) { float s = 0.f; for (int t = 0; t < PRE; ++t) s += pre[t]; ws[ACC_OFF] = s; }

  float acc;
  // GB = A^16 via 4 squarings
  acc = 0.f; for (int q = 0; q < 16; ++q) acc += A[i][q]  * A[q][k];  T1[i][k] = acc; __syncthreads();
  acc = 0.f; for (int q = 0; q < 16; ++q) acc += T1[i][q] * T1[q][k]; T2[i][k] = acc; __syncthreads();
  acc = 0.f; for (int q = 0; q < 16; ++q) acc += T2[i][q] * T2[q][k]; __syncthreads();
  T1[i][k] = acc; __syncthreads();
  acc = 0.f; for (int q = 0; q < 16; ++q) acc += T1[i][q] * T1[q][k];
  GBsh[i][k] = acc; ws[GB_OFF + tid] = acc; __syncthreads();

  // U columns: U[:,15-s] = A^s e15
  if (tid < 16) u[tid] = (tid == 15) ? 1.f : 0.f;
  __syncthreads();
  for (int s = 0; s < 16; ++s) {
    if (tid < 16) {
      Ush[tid][15 - s] = u[tid];
      ws[U_OFF + tid * 16 + (15 - s)] = u[tid];
      float t = 0.f; for (int q = 0; q < 16; ++q) t += A[tid][q] * u[q];
      un[tid] = t;
    }
    __syncthreads();
    if (tid < 16) u[tid] = un[tid];
    __syncthreads();
  }

  // M_j = GB^{15-j} U, j = 15..0 (prefolded so chunk offsets need no layout swap)
  T1[i][k] = Ush[i][k];
  ws[M_OFF + 15 * 256 + tid] = Ush[i][k];
  __syncthreads();
  for (int j = 14; j >= 0; --j) {
    acc = 0.f; for (int q = 0; q < 16; ++q) acc += GBsh[i][q] * T1[q][k];
    __syncthreads();
    T1[i][k] = acc; ws[M_OFF + j * 256 + tid] = acc;
    __syncthreads();
  }

  // GC = GB^16 via 4 squarings
  acc = 0.f; for (int q = 0; q < 16; ++q) acc += GBsh[i][q] * GBsh[q][k]; T1[i][k] = acc; __syncthreads();
  acc = 0.f; for (int q = 0; q < 16; ++q) acc += T1[i][q]   * T1[q][k];   T2[i][k] = acc; __syncthreads();
  acc = 0.f; for (int q = 0; q < 16; ++q) acc += T2[i][q]   * T2[q][k];   __syncthreads();
  T1[i][k] = acc; __syncthreads();
  acc = 0.f; for (int q = 0; q < 16; ++q) acc += T1[i][q]   * T1[q][k];
  ws[GC_OFF + tid] = acc;
}

// ============ phase 2: b_t = y_t - AR_t, LDS-tiled (4096 x 256) ============
__global__ void arima_bstream(const float* __restrict__ series,
                              const float* __restrict__ w_ar,
                              float* __restrict__ ws) {
  __shared__ float ssh[290];
  const int  tid   = threadIdx.x;
  const long basep = (long)blockIdx.x * 256;
  const long ybase = basep + 1;                 // first y index this tile touches
  for (int i2 = tid; i2 < 289; i2 += 256) {
    long idx = ybase + i2;
    ssh[i2] = (idx < S0_LEN) ? series[idx] : 0.f;
  }
  __syncthreads();
  float war[16];
  #pragma unroll
  for (int k = 0; k < 16; ++k) war[k] = w_ar[k];
  long p = basep + tid;                         // recurrence step; t = p + 17
  float b = 0.f;
  if (p < NP_REAL) {
    int l = tid + 16;                           // y_t at ssh diff index l
    float yt = ssh[l + 1] - ssh[l];
    float ar = 0.f;
    #pragma unroll
    for (int k = 0; k < 16; ++k) {              // ar_t = sum_k y[t-16+k] * w_ar[k]
      int m = l - 16 + k;
      ar += war[k] * (ssh[m + 1] - ssh[m]);
    }
    b = yt - ar;
  }
  ws[B_OFF + p] = b;
}

// ====== phase 3: per-chunk offsets v_c = sum_j M_j * B_j  (WMMA, 256 x 32) ======
// One wave handles 16 chunks: chunk index = B-operand column.
__global__ void arima_chunk_v(float* __restrict__ ws) {
  const int lane = threadIdx.x;
  const int half = lane >> 4;        // 0: lanes 0-15, 1: lanes 16-31
  const int lr   = lane & 15;
  const int klo  = half * 2;         // A/B reg0 holds K=0/2, reg1 holds K=1/3
  const int cbase = blockIdx.x * 16;
  const float* __restrict__ bs = ws + B_OFF;
  v8f c = {};
  for (int j = 0; j < 16; ++j) {
    const float* Mj = ws + M_OFF + j * 256;     // row-major 16x16
    #pragma unroll
    for (int kk = 0; kk < 4; ++kk) {
      v2f a, b;
      a[0] = Mj[lr * 16 + kk * 4 + klo];
      a[1] = Mj[lr * 16 + kk * 4 + klo + 1];
      long pb = (long)(cbase + lr) * CSTEPS + j * 16 + kk * 4;  // (K,N)=(row,chunk)
      b[0] = bs[pb + klo];
      b[1] = bs[pb + klo + 1];
      c = wmma4(a, b, c);
    }
  }
  #pragma unroll
  for (int r = 0; r < 8; ++r) {
    int Mrow = r + half * 8;                    // C/D layout: lane owns rows r / r+8
    ws[V_OFF + (cbase + lr) * 16 + Mrow] = c[r];
  }
}

// ====== phase 4: sequential cross-chunk scan s <- GC*s + v_c (1 x 32) ======
__global__ void arima_scan(float* __restrict__ ws) {
  __shared__ float s[16], ns[16];
  const int tid = threadIdx.x;
  float gcrow[16];
  if (tid < 16) {
    #pragma unroll
    for (int q = 0; q < 16; ++q) gcrow[q] = ws[GC_OFF + tid * 16 + q];
    s[tid] = ws[S0_OFF + tid];
  }
  __syncthreads();
  for (int c = 0; c < NCHUNK; ++c) {
    if (tid < 16) {
      ws[SIN_OFF + c * 16 + tid] = s[tid];
      float a = ws[V_OFF + c * 16 + tid];
      #pragma unroll
      for (int q = 0; q < 16; ++q) a += gcrow[q] * s[q];
      ns[tid] = a;
    }
    __syncthreads();
    if (tid < 16) s[tid] = ns[tid];
    __syncthreads();
  }
}

// ====== phase 5: replay E <- GB*E + U*B_j, accumulate err^2 (WMMA, 256 x 32) ======
// Block of 16 steps == state size, so the new state IS the block's 16 errors.
__global__ void arima_errors(float* __restrict__ ws) {
  __shared__ float E[16][17];                   // E[row=step-in-block][col=chunk]
  __shared__ float red[32];
  const int lane = threadIdx.x;
  const int half = lane >> 4, lr = lane & 15, klo = half * 2;
  const int cbase = blockIdx.x * 16;
  v2f agb[4], au[4];
  #pragma unroll
  for (int kk = 0; kk < 4; ++kk) {
    agb[kk][0] = ws[GB_OFF + lr * 16 + kk * 4 + klo];
    agb[kk][1] = ws[GB_OFF + lr * 16 + kk * 4 + klo + 1];
    au[kk][0]  = ws[U_OFF  + lr * 16 + kk * 4 + klo];
    au[kk][1]  = ws[U_OFF  + lr * 16 + kk * 4 + klo + 1];
  }
  #pragma unroll
  for (int r = 0; r < 8; ++r) {                 // seed E with incoming chunk states
    int M = r + half * 8;
    E[M][lr] = ws[SIN_OFF + (cbase + lr) * 16 + M];
  }
  __syncthreads();
  const float* __restrict__ bs = ws + B_OFF;
  float sq = 0.f;
  for (int j = 0; j < 16; ++j) {
    v8f c = {};
    #pragma unroll
    for (int kk = 0; kk < 4; ++kk) {            // GB * E (E read in B-layout from LDS)
      v2f eb;
      eb[0] = E[kk * 4 + klo][lr];
      eb[1] = E[kk * 4 + klo + 1][lr];
      c = wmma4(agb[kk], eb, c);
    }
    #pragma unroll
    for (int kk = 0; kk < 4; ++kk) {            // + U * B_j
      v2f bj;
      long pb = (long)(cbase + lr) * CSTEPS + j * 16 + kk * 4;
      bj[0] = bs[pb + klo];
      bj[1] = bs[pb + klo + 1];
      c = wmma4(au[kk], bj, c);
    }
    __syncthreads();                            // all reads of E done before overwrite
    #pragma unroll
    for (int r = 0; r < 8; ++r) {
      int M = r + half * 8;
      E[M][lr] = c[r];
      long p = (long)(cbase + lr) * CSTEPS + j * 16 + M;
      if (p < NP_REAL) sq += c[r] * c[r];       // mask padded tail steps
    }
    __syncthreads();
  }
  red[lane] = sq;
  __syncthreads();
  if (lane == 0) {
    float t = 0.f;
    for (int q = 0; q < 32; ++q) t += red[q];
    atomicAdd(&ws[ACC_OFF], t);
  }
}

// ================= phase 6: finalize mean =================
__global__ void arima_final(const float* __restrict__ ws, float* __restrict__ out) {
  if (threadIdx.x == 0 && blockIdx.x == 0) out[0] = ws[ACC_OFF] / (float)S_LEN;
}

extern "C" void kernel_launch(void* const* d_in, const int* in_sizes, int n_in,
                              void* d_out, int out_size, void* d_ws, size_t ws_size,
                              hipStream_t stream) {
  (void)in_sizes; (void)n_in; (void)out_size; (void)ws_size;
  const float* series = (const float*)d_in[0];
  const float* w_ar   = (const float*)d_in[1];
  const float* w_ma   = (const float*)d_in[2];
  float* out = (float*)d_out;
  float* ws  = (float*)d_ws;                    // needs ~4.9 MB, well under harness scratch

  arima_setup  <<<1,            256, 0, stream>>>(series, w_ma, ws);
  arima_bstream<<<NP / 256,     256, 0, stream>>>(series, w_ar, ws);
  arima_chunk_v<<<NCHUNK / 16,   32, 0, stream>>>(ws);
  arima_scan   <<<1,             32, 0, stream>>>(ws);
  arima_errors <<<NCHUNK / 16,   32, 0, stream>>>(ws);
  arima_final  <<<1,              1, 0, stream>>>(ws, out);
}